// fusionLossFunc_improved_20950850469966
// MI455X (gfx1250) — compile-verified
//
#include <hip/hip_runtime.h>
#include <hip/hip_bf16.h>

typedef float v2f __attribute__((ext_vector_type(2)));
typedef float v4f __attribute__((ext_vector_type(4)));
typedef float v8f __attribute__((ext_vector_type(8)));

#define LDIM 72
#define HDIM 96
#define WDIM 96
#define RAD2 400
#define IMG  (LDIM * HDIM * WDIM)   /* 663552 elements per (b,lm) image */
#define NF4  (IMG / 4)              /* 165888 float4 per image */
#define BPI  81                     /* blocks per image: 81*256*8*4 == IMG exactly */
#define TPB  256

// Full 32-lane wave sum using the f32 WMMA unit:
//   A (16x4): A[m][0] = s_m (lanes 0-15, VGPR0), A[m][2] = s_{m+16} (lanes 16-31, VGPR0),
//             VGPR1 (K=1,3) = 0.  B (4x16) = ones.
//   => D[m][n] = s_m + s_{m+16} for every n.
//   Sum a lane's 8 C/D VGPRs (rows m..m+7 of its half) + shfl_xor(16) => total in all lanes.
// Caller must guarantee EXEC is all 1s (no divergence) per WMMA restrictions.
__device__ __forceinline__ float wave_reduce_wmma(float s) {
    v2f a; a.x = s;   a.y = 0.0f;
    v2f b; b.x = 1.0f; b.y = 1.0f;
    v8f c = {};
    c = __builtin_amdgcn_wmma_f32_16x16x4_f32(
            /*neg_a=*/false, a, /*neg_b=*/false, b,
            /*c_mod=*/(short)0, c, /*reuse_a=*/false, /*reuse_b=*/false);
    float t = ((c[0] + c[1]) + (c[2] + c[3])) + ((c[4] + c[5]) + (c[6] + c[7]));
    t += __shfl_xor(t, 16, 32);   // combine low-half / high-half row sums
    return t;
}

__global__ __launch_bounds__(TPB) void bce_sphere_partial(
        const float* __restrict__ fm,    // (8,14,72,96,96)
        const float* __restrict__ lms,   // (8,14,3)
        float* __restrict__ partial) {   // (112 * BPI)
    const int img = blockIdx.y;                       // b*14 + lm  (uniform)
    // Scalar (uniform) landmark fetch; round-half-even to match jnp.round.
    const int X = (int)rintf(lms[img * 3 + 0] * (float)(HDIM - 1));
    const int Y = (int)rintf(lms[img * 3 + 1] * (float)(WDIM - 1));
    const int Z = (int)rintf(lms[img * 3 + 2] * (float)(LDIM - 1));

    const v4f* __restrict__ src = (const v4f*)(fm + (size_t)img * IMG);

    float acc = 0.0f;
    for (int i4 = blockIdx.x * TPB + threadIdx.x; i4 < NF4; i4 += BPI * TPB) {
        // Stream once, don't pollute L2: non-temporal 128-bit load.
        v4f v = __builtin_nontemporal_load(&src[i4]);
        const int row = i4 / 24;                 // 24 float4 per y-row (96/4)
        const int y0  = (i4 - row * 24) * 4;     // first y of this float4
        const int x   = row % HDIM;
        const int z   = row / HDIM;
        const int dz  = z - Z;
        const int dx  = x - X;
        const int b2  = dz * dz + dx * dx;
#pragma unroll
        for (int k = 0; k < 4; ++k) {
            const float val = v[k];
            const int   dy  = (y0 + k) - Y;
            const bool  gt  = (b2 + dy * dy) <= RAD2;   // analytic sphere crop
            // BCE-with-logits: max(x,0) - x*gt + log(1 + exp(-|x|))
            const float sp = __logf(1.0f + __expf(-fabsf(val)));
            acc += fmaxf(val, 0.0f) + sp - (gt ? val : 0.0f);
        }
    }

    // Per-wave WMMA reduction (all 8 waves fully active -> EXEC all 1s).
    const float wsum = wave_reduce_wmma(acc);

    __shared__ float red[TPB / 32];
    if ((threadIdx.x & 31) == 0) red[threadIdx.x >> 5] = wsum;
    __syncthreads();
    if (threadIdx.x == 0) {
        float t = 0.0f;
#pragma unroll
        for (int j = 0; j < TPB / 32; ++j) t += red[j];
        partial[blockIdx.y * gridDim.x + blockIdx.x] = t;
    }
}

__global__ __launch_bounds__(TPB) void bce_sphere_final(
        const float* __restrict__ partial, int n, float scale,
        float* __restrict__ out) {
    float acc = 0.0f;
    for (int i = threadIdx.x; i < n; i += TPB) acc += partial[i];

    const float wsum = wave_reduce_wmma(acc);

    __shared__ float red[TPB / 32];
    if ((threadIdx.x & 31) == 0) red[threadIdx.x >> 5] = wsum;
    __syncthreads();
    if (threadIdx.x == 0) {
        float t = 0.0f;
#pragma unroll
        for (int j = 0; j < TPB / 32; ++j) t += red[j];
        out[0] = t * scale;   // overwrite (poison-safe), deterministic
    }
}

extern "C" void kernel_launch(void* const* d_in, const int* in_sizes, int n_in,
                              void* d_out, int out_size, void* d_ws, size_t ws_size,
                              hipStream_t stream) {
    const float* fm  = (const float*)d_in[0];   // feature_map (8,14,72,96,96) f32
    const float* lms = (const float*)d_in[1];   // landmarks   (8,14,3)        f32
    const int nimg = in_sizes[1] / 3;           // B*Lm = 112

    float* partial = (float*)d_ws;              // BPI*nimg floats = 36 KB

    dim3 grid(BPI, nimg);
    bce_sphere_partial<<<grid, TPB, 0, stream>>>(fm, lms, partial);

    const int nPart = BPI * nimg;
    bce_sphere_final<<<1, TPB, 0, stream>>>(partial, nPart,
                                            1.0f / (float)nimg,
                                            (float*)d_out);
}